// HybridLocationPredictor_89352499626172
// MI455X (gfx1250) — compile-verified
//
#include <hip/hip_runtime.h>
#include <math.h>

typedef __attribute__((ext_vector_type(2))) float v2f;
typedef __attribute__((ext_vector_type(8))) float v8f;

#define BB 4096
#define LL 64
#define VV 2001
#define HIDN 88
#define DD 122
#define HSTRIDE 96

// ---------------- Stage 1: encoder + stat scalars, one block per batch row ---
__global__ __launch_bounds__(128) void hlp_stage1(
    const int* __restrict__ locations, const int* __restrict__ users,
    const int* __restrict__ weekdays, const int* __restrict__ start_mins,
    const int* __restrict__ time_diffs, const int* __restrict__ lengths,
    const float* __restrict__ loc_emb, const float* __restrict__ user_emb,
    const float* __restrict__ weekday_emb, const float* __restrict__ time_diff_emb,
    const float* __restrict__ attn_w, const float* __restrict__ proj_W,
    const float* __restrict__ proj_b,
    const float* __restrict__ user_prior, const float* __restrict__ user_trans,
    const float* __restrict__ global_trans, const float* __restrict__ temporal_prior,
    const float* __restrict__ location_prior,
    float* __restrict__ hiddenWS, float* __restrict__ fparams, int* __restrict__ iparams)
{
  const int b = blockIdx.x;
  const int tid = threadIdx.x;
  __shared__ float xs[LL][DD + 2];          // 64 x 124 features
  __shared__ int sloc[LL], susr[LL], swd[LL], ssm[LL], stdf[LL];
  __shared__ float attn_s[LL];
  __shared__ float pooled_s[DD + 2];
  __shared__ float misc[4];
  __shared__ int slen;
  __shared__ float red5[5][128];

  if (tid == 0) slen = lengths[b];
  if (tid < LL) {
    sloc[tid] = locations[b * LL + tid];
    susr[tid] = users[b * LL + tid];
    swd[tid]  = weekdays[b * LL + tid];
    ssm[tid]  = start_mins[b * LL + tid];
    stdf[tid] = time_diffs[b * LL + tid];
  }
  __syncthreads();

  // Build concatenated features x[l][d]
  for (int idx = tid; idx < LL * DD; idx += 128) {
    int l = idx / DD;
    int d = idx - l * DD;
    float v;
    if (d < 88)       v = loc_emb[sloc[l] * 88 + d];
    else if (d < 104) v = user_emb[susr[l] * 16 + (d - 88)];
    else if (d < 112) v = weekday_emb[swd[l] * 8 + (d - 104)];
    else if (d < 114) {
      float theta = (float)ssm[l] * (6.28318530717958647692f / 1440.0f);
      v = (d == 112) ? sinf(theta) : cosf(theta);
    } else            v = time_diff_emb[stdf[l] * 8 + (d - 114)];
    xs[l][d] = v;
  }
  __syncthreads();

  // Attention scores
  if (tid < LL) {
    float s = 0.f;
    for (int d = 0; d < DD; ++d) s += xs[tid][d] * attn_w[d];
    attn_s[tid] = s;
  }
  __syncthreads();
  if (tid == 0) {
    float m = -3.4e38f;
    for (int l = 0; l < slen && l < LL; ++l) m = fmaxf(m, attn_s[l]);
    misc[0] = m;
  }
  __syncthreads();
  if (tid < LL) attn_s[tid] = (tid < slen) ? expf(attn_s[tid] - misc[0]) : 0.f;
  __syncthreads();
  if (tid == 0) {
    float s = 0.f;
    for (int l = 0; l < LL; ++l) s += attn_s[l];
    misc[1] = 1.f / s;
  }
  __syncthreads();
  if (tid < LL) attn_s[tid] *= misc[1];
  __syncthreads();

  // Pooled features
  if (tid < DD) {
    float p = 0.f;
    for (int l = 0; l < LL; ++l) p += attn_s[l] * xs[l][tid];
    pooled_s[tid] = p;
  }
  __syncthreads();

  // hidden = pooled @ proj_W.T + proj_b   (stride 96, zero-padded)
  if (tid < HSTRIDE) {
    float h = 0.f;
    if (tid < HIDN) {
      h = proj_b[tid];
      for (int d = 0; d < DD; ++d) h += pooled_s[d] * proj_W[tid * DD + d];
    }
    hiddenWS[(size_t)b * HSTRIDE + tid] = h;
  }

  // ---- statistical prior scalars ----
  int usr0 = susr[0];
  int has  = slen > 0;
  int idxl = (slen - 1 > 0) ? (slen - 1) : 0;
  int last = has ? sloc[idxl] : 0;
  int wdl  = has ? swd[idxl] : 0;
  int sml  = has ? ssm[idxl] : 0;
  int trow = wdl * 24 + (sml / 60);

  const float* p0 = user_prior     + (size_t)usr0 * VV;
  const float* p1 = user_trans     + ((size_t)usr0 * VV + (size_t)last) * VV;
  const float* p2 = global_trans   + (size_t)last * VV;
  const float* p3 = temporal_prior + (size_t)trow * VV;

  float a0 = 0.f, a1 = 0.f, a2 = 0.f, a3 = 0.f, al = 0.f;
  for (int v = tid; v < VV; v += 128) {
    a0 += p0[v]; a1 += p1[v]; a2 += p2[v]; a3 += p3[v]; al += location_prior[v];
  }
  red5[0][tid] = a0; red5[1][tid] = a1; red5[2][tid] = a2;
  red5[3][tid] = a3; red5[4][tid] = al;
  __syncthreads();

  if (tid == 0) {
    float sums[5];
    for (int k = 0; k < 5; ++k) {
      float s = 0.f;
      for (int t = 0; t < 128; ++t) s += red5[k][t];
      sums[k] = s;
    }
    const float basew[4] = {0.3f, 0.35f, 0.2f, 0.15f};
    float w[4]; float wsum = 0.f;
    for (int k = 0; k < 4; ++k) { w[k] = (sums[k] > 0.01f) ? basew[k] : 0.f; wsum += w[k]; }
    float wloc, S;
    if (wsum > 0.f) {
      float inv = 1.f / wsum; S = 0.f;
      for (int k = 0; k < 4; ++k) { w[k] *= inv; S += w[k] * sums[k]; }
      wloc = 0.f;
    } else {
      for (int k = 0; k < 4; ++k) w[k] = 0.f;
      wloc = 1.f; S = sums[4];
    }
    float invSp = 1.f / (S + 1e-8f);
    float logZ  = logf(S * invSp + (float)VV * 1e-8f);
    float* fp = fparams + (size_t)b * 8;
    fp[0] = w[0]; fp[1] = w[1]; fp[2] = w[2]; fp[3] = w[3];
    fp[4] = wloc; fp[5] = invSp; fp[6] = logZ; fp[7] = 0.f;
    int* ip = iparams + (size_t)b * 4;
    ip[0] = usr0; ip[1] = last; ip[2] = trow; ip[3] = 0;
  }
}

// ---------------- Stage 2: fp32 WMMA GEMM + fused stat epilogue --------------
__global__ __launch_bounds__(128) void hlp_stage2(
    const float* __restrict__ hiddenWS, const float* __restrict__ loc_emb,
    const float* __restrict__ fparams, const int* __restrict__ iparams,
    const float* __restrict__ user_prior, const float* __restrict__ user_trans,
    const float* __restrict__ global_trans, const float* __restrict__ temporal_prior,
    const float* __restrict__ location_prior, const float* __restrict__ ensemble_w,
    float* __restrict__ out)
{
  const int wv    = threadIdx.x >> 5;
  const int lane  = threadIdx.x & 31;
  const int NTILES = (VV + 15) / 16;              // 126
  const int ntile = blockIdx.y * 4 + wv;
  if (ntile >= NTILES) return;                    // wave-uniform exit

  const int mbase = blockIdx.x * 16;
  const int nbase = ntile * 16;
  const int m16   = lane & 15;
  const int khalf = lane >> 4;
  const int vcol  = nbase + m16;
  const int vc    = (vcol < VV) ? vcol : (VV - 1);

  const float* hrow = hiddenWS + (size_t)(mbase + m16) * HSTRIDE;
  const float* lrow = loc_emb  + (size_t)vc * HIDN;

  v8f acc = {0.f, 0.f, 0.f, 0.f, 0.f, 0.f, 0.f, 0.f};
#pragma unroll
  for (int kb = 0; kb < HIDN; kb += 4) {
    int k0 = kb + 2 * khalf;
    v2f a;  a.x  = hrow[k0]; a.y  = hrow[k0 + 1];   // A: 16x4 fp32 fragment
    v2f bm; bm.x = lrow[k0]; bm.y = lrow[k0 + 1];   // B: 4x16 fp32 fragment
    acc = __builtin_amdgcn_wmma_f32_16x16x4_f32(
        /*neg_a=*/false, a, /*neg_b=*/false, bm,
        /*c_mod=*/(short)0, acc, /*reuse_a=*/false, /*reuse_b=*/false);
  }

  const float alpha = 1.f / (1.f + expf(-ensemble_w[0]));
  const float beta  = (1.f - alpha) * 0.5f;
  const bool  vok   = (vcol < VV);
  const float lp    = location_prior[vc];

#pragma unroll
  for (int r = 0; r < 8; ++r) {
    int brow = mbase + r + 8 * khalf;               // C/D layout: M = r + 8*(lane>>4)
    const float* fp = fparams + (size_t)brow * 8;
    const int*   ip = iparams + (size_t)brow * 4;
    int usr0 = ip[0], last = ip[1], trow = ip[2];
    float pv0 = user_prior[(size_t)usr0 * VV + vc];
    float pv1 = user_trans[((size_t)usr0 * VV + (size_t)last) * VV + vc];
    float pv2 = global_trans[(size_t)last * VV + vc];
    float pv3 = temporal_prior[(size_t)trow * VV + vc];
    float comb = fp[0] * pv0 + fp[1] * pv1 + fp[2] * pv2 + fp[3] * pv3 + fp[4] * lp;
    float stat = logf(comb * fp[5] + 1e-8f) - fp[6];
    if (vok) out[(size_t)brow * VV + vcol] = alpha * acc[r] + beta * stat;
  }
}

// ---------------- Launch -----------------------------------------------------
extern "C" void kernel_launch(void* const* d_in, const int* in_sizes, int n_in,
                              void* d_out, int out_size, void* d_ws, size_t ws_size,
                              hipStream_t stream) {
  (void)in_sizes; (void)n_in; (void)out_size; (void)ws_size;
  const int*   locations      = (const int*)d_in[0];
  const int*   users          = (const int*)d_in[1];
  const int*   weekdays       = (const int*)d_in[2];
  const int*   start_mins     = (const int*)d_in[3];
  /* durations d_in[4] unused */
  const int*   time_diffs     = (const int*)d_in[5];
  const int*   lengths        = (const int*)d_in[6];
  const float* loc_emb        = (const float*)d_in[7];
  const float* user_emb       = (const float*)d_in[8];
  const float* weekday_emb    = (const float*)d_in[9];
  const float* time_diff_emb  = (const float*)d_in[10];
  const float* attn_w         = (const float*)d_in[11];
  const float* proj_W         = (const float*)d_in[12];
  const float* proj_b         = (const float*)d_in[13];
  const float* ensemble_w     = (const float*)d_in[14];
  const float* user_prior     = (const float*)d_in[15];
  const float* user_trans     = (const float*)d_in[16];
  const float* global_trans   = (const float*)d_in[17];
  const float* temporal_prior = (const float*)d_in[18];
  const float* location_prior = (const float*)d_in[19];
  float* out = (float*)d_out;

  float* hiddenWS = (float*)d_ws;                       // 4096*96 floats
  float* fparams  = hiddenWS + (size_t)BB * HSTRIDE;    // 4096*8 floats
  int*   iparams  = (int*)(fparams + (size_t)BB * 8);   // 4096*4 ints

  hlp_stage1<<<BB, 128, 0, stream>>>(
      locations, users, weekdays, start_mins, time_diffs, lengths,
      loc_emb, user_emb, weekday_emb, time_diff_emb,
      attn_w, proj_W, proj_b,
      user_prior, user_trans, global_trans, temporal_prior, location_prior,
      hiddenWS, fparams, iparams);

  dim3 g2(BB / 16, ((VV + 15) / 16 + 3) / 4);           // (256, 32)
  hlp_stage2<<<g2, 128, 0, stream>>>(
      hiddenWS, loc_emb, fparams, iparams,
      user_prior, user_trans, global_trans, temporal_prior,
      location_prior, ensemble_w, out);
}